// GNNSage_46943992545895
// MI455X (gfx1250) — compile-verified
//
#include <hip/hip_runtime.h>

typedef __attribute__((ext_vector_type(2))) float v2f;
typedef __attribute__((ext_vector_type(8))) float v8f;

static constexpr int NNODES = 50000;
static constexpr int NEDGES = 800000;
static constexpr int FEAT   = 64;   // IN_FEATS == HIDDEN == 64

// ---------------- utility: zero a float buffer ----------------
__global__ void gnn_zero_f32(float* __restrict__ p, int n) {
  int i = blockIdx.x * blockDim.x + threadIdx.x;
  if (i < n) p[i] = 0.0f;
}

// ---------------- degree count (float so divide is exact later) -------------
__global__ void gnn_degree(const int* __restrict__ dst, float* __restrict__ deg) {
  int e = blockIdx.x * blockDim.x + threadIdx.x;
  if (e < NEDGES) atomicAdd(&deg[dst[e]], 1.0f);
}

// ---------------- in-place deg -> 1/max(deg,1) ------------------------------
__global__ void gnn_recip(float* __restrict__ deg, int n) {
  int i = blockIdx.x * blockDim.x + threadIdx.x;
  if (i < n) deg[i] = 1.0f / fmaxf(deg[i], 1.0f);
}

// ---------------- feature scatter-add: agg[dst] += x[src] -------------------
// 16 threads per edge, each handles 4 contiguous features (float4 read).
__global__ void gnn_scatter(const float* __restrict__ x,
                            const int* __restrict__ src,
                            const int* __restrict__ dst,
                            float* __restrict__ agg) {
  int t = blockIdx.x * blockDim.x + threadIdx.x;
  int e = t >> 4;
  int f = (t & 15) * 4;
  if (e >= NEDGES) return;
  int s = src[e];
  int d = dst[e];
  float4 v = *reinterpret_cast<const float4*>(x + s * FEAT + f);
  float* o = agg + d * FEAT + f;
  atomicAdd(o + 0, v.x);
  atomicAdd(o + 1, v.y);
  atomicAdd(o + 2, v.z);
  atomicAdd(o + 3, v.w);
}

// ---------------- fused SAGE layer: out = (agg*invdeg)@Wl + bl + x@Wr -------
// One wave per (16-node row block, 16-col N-tile). K=64 in 16 steps of
// V_WMMA_F32_16X16X4_F32. Layouts per CDNA5 ISA:
//   A 16x4 : lanes 0-15 -> {K0,K1}, lanes 16-31 -> {K2,K3}; M = lane&15
//   B 4x16 : v0 -> rows {K0|K2}, v1 -> rows {K1|K3}; N = lane&15
//   C 16x16: VGPR r -> M = r (lanes 0-15) / r+8 (lanes 16-31); N = lane&15
// For NOUT=8 the B column index is clamped (col & 7): D columns 8..15 become
// garbage but are never stored, so all loads stay unconditional (EXEC all-1s).
template <int NOUT, int NWAVES>
__global__ __launch_bounds__(NWAVES * 32)
void gnn_sage_layer_wmma(const float* __restrict__ agg,
                         const float* __restrict__ invdeg,
                         const float* __restrict__ xr,
                         const float* __restrict__ Wl,
                         const float* __restrict__ bl,
                         const float* __restrict__ Wr,
                         float* __restrict__ out) {
  const int lane  = threadIdx.x & 31;
  const int wave  = threadIdx.x >> 5;
  const int row0  = blockIdx.x * 16;
  const int n0    = wave * 16;
  const int nn    = lane & 15;            // N (for B/C/D) and M (for A)
  const int khalf = (lane >> 4) * 2;      // 0 for lanes 0-15, 2 for 16-31
  const int col   = n0 + nn;
  const int bcol  = col & (NOUT - 1);     // clamped, always in-range

  const float invd = invdeg[row0 + nn];
  const float* __restrict__ arow = agg + (row0 + nn) * FEAT;
  const float* __restrict__ xrow = xr  + (row0 + nn) * FEAT;

  v8f c = {};

  // (agg * invdeg) @ Wl
#pragma unroll
  for (int ks = 0; ks < FEAT / 4; ++ks) {
    const int k0 = ks * 4 + khalf;
    v2f a, b;
    a.x = arow[k0]     * invd;
    a.y = arow[k0 + 1] * invd;
    b.x = Wl[k0 * NOUT + bcol];
    b.y = Wl[(k0 + 1) * NOUT + bcol];
    c = __builtin_amdgcn_wmma_f32_16x16x4_f32(false, a, false, b,
                                              (short)0, c, false, false);
  }
  // + x @ Wr   (same accumulator)
#pragma unroll
  for (int ks = 0; ks < FEAT / 4; ++ks) {
    const int k0 = ks * 4 + khalf;
    v2f a, b;
    a.x = xrow[k0];
    a.y = xrow[k0 + 1];
    b.x = Wr[k0 * NOUT + bcol];
    b.y = Wr[(k0 + 1) * NOUT + bcol];
    c = __builtin_amdgcn_wmma_f32_16x16x4_f32(false, a, false, b,
                                              (short)0, c, false, false);
  }

  if (col < NOUT) {
    const float bias  = bl[col];
    const int   mbase = (lane >> 4) * 8;
#pragma unroll
    for (int r = 0; r < 8; ++r) {
      const int row = row0 + mbase + r;
      out[row * NOUT + col] = c[r] + bias;
    }
  }
}

extern "C" void kernel_launch(void* const* d_in, const int* in_sizes, int n_in,
                              void* d_out, int out_size, void* d_ws, size_t ws_size,
                              hipStream_t stream) {
  const float* embeds = (const float*)d_in[0];
  const int*   eidx   = (const int*)d_in[1];
  const float* W_l1   = (const float*)d_in[2];
  const float* b_l1   = (const float*)d_in[3];
  const float* W_r1   = (const float*)d_in[4];
  const float* W_l2   = (const float*)d_in[5];
  const float* b_l2   = (const float*)d_in[6];
  const float* W_r2   = (const float*)d_in[7];
  float* out = (float*)d_out;

  const int* src = eidx;            // edge_index[0]
  const int* dst = eidx + NEDGES;   // edge_index[1]

  float* ws  = (float*)d_ws;
  float* deg = ws;                                 // NNODES (becomes invdeg)
  float* agg = ws + NNODES;                        // NNODES*FEAT (reused)
  float* h   = agg + (size_t)NNODES * FEAT;        // NNODES*FEAT

  // --- layer 1 aggregation ---
  {
    int n = NNODES * (FEAT + 1);                   // zero deg + agg together
    gnn_zero_f32<<<(n + 255) / 256, 256, 0, stream>>>(ws, n);
  }
  gnn_degree<<<(NEDGES + 255) / 256, 256, 0, stream>>>(dst, deg);
  gnn_recip<<<(NNODES + 255) / 256, 256, 0, stream>>>(deg, NNODES);
  gnn_scatter<<<(NEDGES * 16 + 255) / 256, 256, 0, stream>>>(embeds, src, dst, agg);

  // --- layer 1 fused GEMMs (4 N-tiles of 16 cols -> 4 waves / block) ---
  gnn_sage_layer_wmma<64, 4><<<NNODES / 16, 128, 0, stream>>>(
      agg, deg, embeds, W_l1, b_l1, W_r1, h);

  // --- layer 2 aggregation (reuse agg) ---
  {
    int n = NNODES * FEAT;
    gnn_zero_f32<<<(n + 255) / 256, 256, 0, stream>>>(agg, n);
  }
  gnn_scatter<<<(NEDGES * 16 + 255) / 256, 256, 0, stream>>>(h, src, dst, agg);

  // --- layer 2 fused GEMMs (8 valid cols, padded to one 16-col tile) ---
  gnn_sage_layer_wmma<8, 1><<<NNODES / 16, 32, 0, stream>>>(
      agg, deg, h, W_l2, b_l2, W_r2, out);
}